// LennardJonesModel_30176440222240
// MI455X (gfx1250) — compile-verified
//
#include <hip/hip_runtime.h>
#include <stdint.h>

typedef int   v4i __attribute__((ext_vector_type(4)));
typedef float v4f __attribute__((ext_vector_type(4)));

#define CUTOFF2 6.25f   // (2.5 * sigma)^2, sigma = 1

struct Cell {
  float c00, c01, c02, c10, c11, c12, c20, c21, c22;
};

// One pair: accumulate energy into esum, gated scatter-atomics for forces.
__device__ __forceinline__ void lj_pair_math(const Cell& C,
                                             int i, int j,
                                             float pjx, float pjy, float pjz,
                                             float pix, float piy, float piz,
                                             float sx, float sy, float sz,
                                             float* __restrict__ forces,
                                             float& esum) {
  // dr = r_j - r_i + shifts @ cell   (row-vector times matrix)
  float dx = pjx - pix + sx * C.c00 + sy * C.c10 + sz * C.c20;
  float dy = pjy - piy + sx * C.c01 + sy * C.c11 + sz * C.c21;
  float dz = pjz - piz + sx * C.c02 + sy * C.c12 + sz * C.c22;
  float r2 = dx * dx + dy * dy + dz * dz;
  if (r2 < CUTOFF2) {                           // ~15% of pairs; gates the atomics
    float inv_r2 = __builtin_amdgcn_rcpf(r2);   // v_rcp_f32; no sqrt/div needed
    float sr6  = inv_r2 * inv_r2 * inv_r2;      // (sigma/r)^6, sigma = 1
    float sr12 = sr6 * sr6;
    esum += 4.0f * (sr12 - sr6);
    // force_vec = 24*(2*sr12 - sr6)/r^2 * dr   (== (pair_forces / r) * dr)
    float fs = 24.0f * (2.0f * sr12 - sr6) * inv_r2;
    float fx = fs * dx, fy = fs * dy, fz = fs * dz;
    float* fi = forces + 3 * (long)i;
    float* fj = forces + 3 * (long)j;
    unsafeAtomicAdd(fi + 0, -fx);
    unsafeAtomicAdd(fi + 1, -fy);
    unsafeAtomicAdd(fi + 2, -fz);
    unsafeAtomicAdd(fj + 0,  fx);
    unsafeAtomicAdd(fj + 1,  fy);
    unsafeAtomicAdd(fj + 2,  fz);
  }
}

__global__ void __launch_bounds__(256) lj_pairs_kernel(
    const float* __restrict__ pos,     // [N_ATOMS*3]
    const float* __restrict__ cell,    // [9]
    const int*   __restrict__ map,     // [2*P] : i's then j's
    const float* __restrict__ shifts,  // [P*3]
    float* __restrict__ out,           // [1 + 3*N_ATOMS] : energy, forces
    int P) {
  // 48 B of shift data per thread, staged via async global->LDS copies.
  __shared__ __align__(16) float sshift[256 * 12];
  __shared__ float wsum[8];

  // Uniform address + constant indices -> scalar (SMEM) loads of the cell.
  Cell C;
  C.c00 = cell[0]; C.c01 = cell[1]; C.c02 = cell[2];
  C.c10 = cell[3]; C.c11 = cell[4]; C.c12 = cell[5];
  C.c20 = cell[6]; C.c21 = cell[7]; C.c22 = cell[8];

  float* energy = out;
  float* forces = out + 1;

  int t = blockIdx.x * blockDim.x + threadIdx.x;
  long base = 4L * t;
  float esum = 0.0f;

  if (base + 3 < (long)P) {
    // Mapping quads: needed first (they feed the gathers) -> direct NT b128.
    v4i iv = __builtin_nontemporal_load((const v4i*)map + t);
    v4i jv = __builtin_nontemporal_load((const v4i*)(map + P) + t);

    // Shifts: consumed last -> stream 48 B/thread into LDS asynchronously
    // (ASYNCcnt path, NT so the 201 MB stream doesn't evict L2-resident
    // positions). offset:N advances both the global and LDS addresses.
    uint32_t laddr = (uint32_t)(uintptr_t)&sshift[threadIdx.x * 12];
    uint32_t voff  = 48u * (uint32_t)t;
    asm volatile(
        "global_load_async_to_lds_b128 %0, %1, %2 offset:0 th:TH_LOAD_NT\n\t"
        "global_load_async_to_lds_b128 %0, %1, %2 offset:16 th:TH_LOAD_NT\n\t"
        "global_load_async_to_lds_b128 %0, %1, %2 offset:32 th:TH_LOAD_NT"
        :: "v"(laddr), "v"(voff), "s"((uint64_t)(uintptr_t)shifts)
        : "memory");

    // Random position gathers (L2-resident, 3 MB) overlap the async stream.
    float pix[4], piy[4], piz[4], pjx[4], pjy[4], pjz[4];
#pragma unroll
    for (int k = 0; k < 4; ++k) {
      const float* pi = pos + 3 * (long)iv[k];
      const float* pj = pos + 3 * (long)jv[k];
      pix[k] = pi[0]; piy[k] = pi[1]; piz[k] = pi[2];
      pjx[k] = pj[0]; pjy[k] = pj[1]; pjz[k] = pj[2];
    }

    // Now consume the staged shifts.
    asm volatile("s_wait_asynccnt 0" ::: "memory");
    const v4f* sb = (const v4f*)&sshift[threadIdx.x * 12];
    v4f s0 = sb[0], s1 = sb[1], s2 = sb[2];
    float sx[4] = {s0[0], s0[3], s1[2], s2[1]};
    float sy[4] = {s0[1], s1[0], s1[3], s2[2]};
    float sz[4] = {s0[2], s1[1], s2[0], s2[3]};

#pragma unroll
    for (int k = 0; k < 4; ++k)
      lj_pair_math(C, iv[k], jv[k], pjx[k], pjy[k], pjz[k],
                   pix[k], piy[k], piz[k], sx[k], sy[k], sz[k], forces, esum);
  } else if (base < (long)P) {
    // Scalar tail (P % 4 != 0 safety; unused for P = 2^24).
    for (long p = base; p < (long)P; ++p) {
      int i = map[p], j = map[(long)P + p];
      const float* pi = pos + 3 * (long)i;
      const float* pj = pos + 3 * (long)j;
      lj_pair_math(C, i, j, pj[0], pj[1], pj[2], pi[0], pi[1], pi[2],
                   shifts[3 * p], shifts[3 * p + 1], shifts[3 * p + 2],
                   forces, esum);
    }
  }

  // Energy reduction: wave32 butterfly -> LDS across 8 waves -> 1 atomic/block.
#pragma unroll
  for (int o = 16; o > 0; o >>= 1) esum += __shfl_xor(esum, o, 32);

  int lane = threadIdx.x & 31;
  int wave = threadIdx.x >> 5;
  if (lane == 0) wsum[wave] = esum;
  __syncthreads();
  if (threadIdx.x == 0) {
    float b = 0.0f;
#pragma unroll
    for (int w = 0; w < 8; ++w) b += wsum[w];
    unsafeAtomicAdd(energy, 0.5f * b);   // reference: 0.5 * sum(pair_energies)
  }
}

__global__ void __launch_bounds__(256) lj_zero_kernel(float* __restrict__ out, int n) {
  int t = blockIdx.x * blockDim.x + threadIdx.x;
  if (t < n) out[t] = 0.0f;
}

extern "C" void kernel_launch(void* const* d_in, const int* in_sizes, int n_in,
                              void* d_out, int out_size, void* d_ws, size_t ws_size,
                              hipStream_t stream) {
  const float* pos    = (const float*)d_in[0];  // positions [N_ATOMS,3]
  const float* cell   = (const float*)d_in[1];  // cell [3,3]
  const int*   map    = (const int*)d_in[2];    // mapping [2,P]
  const float* shifts = (const float*)d_in[3];  // shifts [P,3]
  float* out = (float*)d_out;                   // [0]=energy, [1..]=forces

  int P = in_sizes[2] / 2;

  // d_out is poisoned and we accumulate with atomics: zero it every call.
  int nz = out_size;
  lj_zero_kernel<<<(nz + 255) / 256, 256, 0, stream>>>(out, nz);

  int nq = (P + 3) / 4;                          // quads of pairs
  int blocks = (nq + 255) / 256;
  lj_pairs_kernel<<<blocks, 256, 0, stream>>>(pos, cell, map, shifts, out, P);
}